// FraudGAT_40355512713798
// MI455X (gfx1250) — compile-verified
//
#include <hip/hip_runtime.h>
#include <math.h>

typedef __attribute__((ext_vector_type(2))) float v2f;
typedef __attribute__((ext_vector_type(8))) float v8f;

#define NEG_SLOPE 0.2f
#define EPS_F 1e-16f

// -------- float atomic max via int punning (lowers to global_atomic_max/min) --------
__device__ inline void atomicMaxFloat(float* addr, float val) {
    if (val >= 0.0f) {
        atomicMax((int*)addr, __float_as_int(val));
    } else {
        atomicMin((unsigned int*)addr, __float_as_uint(val));
    }
}

// ---------------- init workspace ----------------
__global__ void init_ws(float* m1, float* den1, float* out1,
                        float* m2, float* den2, float* out2, int N) {
    int i = blockIdx.x * blockDim.x + threadIdx.x;
    if (i < N * 128) out1[i] = 0.0f;
    if (i < N * 4) { m1[i] = -INFINITY; den1[i] = 0.0f; }
    if (i < N * 2) out2[i] = 0.0f;
    if (i < N)     { m2[i] = -INFINITY; den2[i] = 0.0f; }
}

// ---------------- layer-1 GEMM: H1[N,128] = X[N,128] @ W1[128,128] via f32 WMMA ----------------
// block = 256 threads = 8 waves; each block does a 16-row stripe; wave w does cols [16w,16w+16)
__global__ __launch_bounds__(256) void gemm_wmma_f32(const float* __restrict__ X,
                                                     const float* __restrict__ W,
                                                     float* __restrict__ H, int N) {
    const int rowBase = blockIdx.x * 16;
    const int wave = threadIdx.x >> 5;
    const int lane = threadIdx.x & 31;
    const int colBase = wave * 16;
    const int half = lane >> 4;      // selects K pair {0,1} vs {2,3}
    const int mn = lane & 15;        // M index for A, N index for B/C/D

    v8f c = {};
    const float* xrow = X + (size_t)(rowBase + mn) * 128;
    #pragma unroll 4
    for (int k = 0; k < 128; k += 4) {
        const int ka = k + 2 * half;
        v2f a;
        a.x = xrow[ka];
        a.y = xrow[ka + 1];
        v2f b;
        b.x = W[(size_t)ka * 128 + colBase + mn];
        b.y = W[(size_t)(ka + 1) * 128 + colBase + mn];
        c = __builtin_amdgcn_wmma_f32_16x16x4_f32(false, a, false, b,
                                                  (short)0, c, false, false);
    }
    #pragma unroll
    for (int r = 0; r < 8; ++r) {
        H[(size_t)(rowBase + r + 8 * half) * 128 + colBase + mn] = c[r];
    }
}

// ---------------- layer-1 attention scalars: a_src/a_dst [N,4] ----------------
__global__ void attn_l1(const float* __restrict__ h1,
                        const float* __restrict__ att_s,
                        const float* __restrict__ att_d,
                        float* __restrict__ as1, float* __restrict__ ad1, int N) {
    int t = blockIdx.x * blockDim.x + threadIdx.x;
    if (t >= N * 4) return;
    int n = t >> 2, h = t & 3;
    const float4* hp = (const float4*)(h1 + (size_t)n * 128 + h * 32);
    const float4* sp = (const float4*)(att_s + h * 32);
    const float4* dp = (const float4*)(att_d + h * 32);
    float s = 0.0f, d = 0.0f;
    #pragma unroll
    for (int j = 0; j < 8; ++j) {
        float4 hv = hp[j], sv = sp[j], dv = dp[j];
        s += hv.x * sv.x + hv.y * sv.y + hv.z * sv.z + hv.w * sv.w;
        d += hv.x * dv.x + hv.y * dv.y + hv.z * dv.z + hv.w * dv.w;
    }
    as1[t] = s;
    ad1[t] = d;
}

// ---------------- layer-1 softmax pass 1: segment max ----------------
__global__ void edge_max_l1(const int* __restrict__ ei, int E, int N,
                            const float* __restrict__ as1, const float* __restrict__ ad1,
                            float* __restrict__ m1) {
    int t = blockIdx.x * blockDim.x + threadIdx.x;
    int tot = (E + N) * 4;
    if (t >= tot) return;
    int e = t >> 2, h = t & 3;
    int src, dst;
    if (e < E) { src = ei[e]; dst = ei[E + e]; } else { src = dst = e - E; }
    float v = as1[src * 4 + h] + ad1[dst * 4 + h];
    v = v >= 0.0f ? v : NEG_SLOPE * v;
    atomicMaxFloat(&m1[dst * 4 + h], v);
}

// ---------------- layer-1 softmax pass 2: segment sum of exp ----------------
__global__ void edge_sum_l1(const int* __restrict__ ei, int E, int N,
                            const float* __restrict__ as1, const float* __restrict__ ad1,
                            const float* __restrict__ m1, float* __restrict__ den1) {
    int t = blockIdx.x * blockDim.x + threadIdx.x;
    int tot = (E + N) * 4;
    if (t >= tot) return;
    int e = t >> 2, h = t & 3;
    int src, dst;
    if (e < E) { src = ei[e]; dst = ei[E + e]; } else { src = dst = e - E; }
    float v = as1[src * 4 + h] + ad1[dst * 4 + h];
    v = v >= 0.0f ? v : NEG_SLOPE * v;
    float mm = m1[dst * 4 + h];
    if (!isfinite(mm)) mm = 0.0f;
    atomicAdd(&den1[dst * 4 + h], expf(v - mm));
}

// ---------------- layer-1 aggregation: wave32 per edge, float4 per lane ----------------
__global__ __launch_bounds__(256) void edge_agg_l1(const int* __restrict__ ei, int E, int N,
                                                   const float* __restrict__ as1,
                                                   const float* __restrict__ ad1,
                                                   const float* __restrict__ m1,
                                                   const float* __restrict__ den1,
                                                   const float* __restrict__ h1,
                                                   float* __restrict__ out1) {
    int e = blockIdx.x * 8 + (threadIdx.x >> 5);
    if (e >= E + N) return;
    int lane = threadIdx.x & 31;
    int src, dst;
    if (e < E) { src = ei[e]; dst = ei[E + e]; } else { src = dst = e - E; }
    int h = lane >> 3;                    // lane*4 channels -> head = (lane*4)/32
    float v = as1[src * 4 + h] + ad1[dst * 4 + h];
    v = v >= 0.0f ? v : NEG_SLOPE * v;
    float mm = m1[dst * 4 + h];
    if (!isfinite(mm)) mm = 0.0f;
    float alpha = expf(v - mm) / (den1[dst * 4 + h] + EPS_F);
    const float4* hs = (const float4*)(h1 + (size_t)src * 128);
    float4 hv = hs[lane];
    float* ob = out1 + (size_t)dst * 128 + lane * 4;
    atomicAdd(ob + 0, alpha * hv.x);
    atomicAdd(ob + 1, alpha * hv.y);
    atomicAdd(ob + 2, alpha * hv.z);
    atomicAdd(ob + 3, alpha * hv.w);
}

// ---------------- bias + ELU (in-place on out1) ----------------
__global__ void bias_elu(float* __restrict__ out1, const float* __restrict__ b1, int N) {
    int i = blockIdx.x * blockDim.x + threadIdx.x;
    if (i >= N * 128) return;
    float v = out1[i] + b1[i & 127];
    out1[i] = v > 0.0f ? v : expf(v) - 1.0f;
}

// ---------------- layer-2 matvec + attention scalars: wave32 per node ----------------
__global__ __launch_bounds__(256) void l2_matvec(const float* __restrict__ hf,
                                                 const float* __restrict__ W2,
                                                 const float* __restrict__ att_s2,
                                                 const float* __restrict__ att_d2,
                                                 float* __restrict__ h2,
                                                 float* __restrict__ as2,
                                                 float* __restrict__ ad2, int N) {
    int n = blockIdx.x * 8 + (threadIdx.x >> 5);
    if (n >= N) return;
    int lane = threadIdx.x & 31;
    const float4* hp = (const float4*)(hf + (size_t)n * 128);
    float4 hv = hp[lane];
    int c = lane * 4;
    float s0 = hv.x * W2[(c + 0) * 2]     + hv.y * W2[(c + 1) * 2]
             + hv.z * W2[(c + 2) * 2]     + hv.w * W2[(c + 3) * 2];
    float s1 = hv.x * W2[(c + 0) * 2 + 1] + hv.y * W2[(c + 1) * 2 + 1]
             + hv.z * W2[(c + 2) * 2 + 1] + hv.w * W2[(c + 3) * 2 + 1];
    #pragma unroll
    for (int off = 16; off >= 1; off >>= 1) {
        s0 += __shfl_xor(s0, off, 32);
        s1 += __shfl_xor(s1, off, 32);
    }
    if (lane == 0) {
        h2[n * 2 + 0] = s0;
        h2[n * 2 + 1] = s1;
        as2[n] = s0 * att_s2[0] + s1 * att_s2[1];
        ad2[n] = s0 * att_d2[0] + s1 * att_d2[1];
    }
}

// ---------------- layer-2 edge passes (1 head) ----------------
__global__ void edge_max_l2(const int* __restrict__ ei, int E, int N,
                            const float* __restrict__ as2, const float* __restrict__ ad2,
                            float* __restrict__ m2) {
    int e = blockIdx.x * blockDim.x + threadIdx.x;
    if (e >= E + N) return;
    int src, dst;
    if (e < E) { src = ei[e]; dst = ei[E + e]; } else { src = dst = e - E; }
    float v = as2[src] + ad2[dst];
    v = v >= 0.0f ? v : NEG_SLOPE * v;
    atomicMaxFloat(&m2[dst], v);
}

__global__ void edge_sum_l2(const int* __restrict__ ei, int E, int N,
                            const float* __restrict__ as2, const float* __restrict__ ad2,
                            const float* __restrict__ m2, float* __restrict__ den2) {
    int e = blockIdx.x * blockDim.x + threadIdx.x;
    if (e >= E + N) return;
    int src, dst;
    if (e < E) { src = ei[e]; dst = ei[E + e]; } else { src = dst = e - E; }
    float v = as2[src] + ad2[dst];
    v = v >= 0.0f ? v : NEG_SLOPE * v;
    float mm = m2[dst];
    if (!isfinite(mm)) mm = 0.0f;
    atomicAdd(&den2[dst], expf(v - mm));
}

__global__ void edge_agg_l2(const int* __restrict__ ei, int E, int N,
                            const float* __restrict__ as2, const float* __restrict__ ad2,
                            const float* __restrict__ m2, const float* __restrict__ den2,
                            const float* __restrict__ h2, float* __restrict__ out2) {
    int e = blockIdx.x * blockDim.x + threadIdx.x;
    if (e >= E + N) return;
    int src, dst;
    if (e < E) { src = ei[e]; dst = ei[E + e]; } else { src = dst = e - E; }
    float v = as2[src] + ad2[dst];
    v = v >= 0.0f ? v : NEG_SLOPE * v;
    float mm = m2[dst];
    if (!isfinite(mm)) mm = 0.0f;
    float alpha = expf(v - mm) / (den2[dst] + EPS_F);
    atomicAdd(&out2[dst * 2 + 0], alpha * h2[src * 2 + 0]);
    atomicAdd(&out2[dst * 2 + 1], alpha * h2[src * 2 + 1]);
}

// ---------------- bias + log_softmax over 2 classes ----------------
__global__ void final_logsoftmax(const float* __restrict__ out2,
                                 const float* __restrict__ b2,
                                 float* __restrict__ y, int N) {
    int n = blockIdx.x * blockDim.x + threadIdx.x;
    if (n >= N) return;
    float z0 = out2[n * 2 + 0] + b2[0];
    float z1 = out2[n * 2 + 1] + b2[1];
    float mx = fmaxf(z0, z1);
    float lse = mx + logf(expf(z0 - mx) + expf(z1 - mx));
    y[n * 2 + 0] = z0 - lse;
    y[n * 2 + 1] = z1 - lse;
}

extern "C" void kernel_launch(void* const* d_in, const int* in_sizes, int n_in,
                              void* d_out, int out_size, void* d_ws, size_t ws_size,
                              hipStream_t stream) {
    const float* x        = (const float*)d_in[0];
    const int*   ei       = (const int*)d_in[1];
    const float* W1       = (const float*)d_in[2];
    const float* att_src1 = (const float*)d_in[3];
    const float* att_dst1 = (const float*)d_in[4];
    const float* b1       = (const float*)d_in[5];
    const float* W2       = (const float*)d_in[6];
    const float* att_src2 = (const float*)d_in[7];
    const float* att_dst2 = (const float*)d_in[8];
    const float* b2       = (const float*)d_in[9];
    float* y = (float*)d_out;

    const int N = in_sizes[0] / 128;   // 50000
    const int E = in_sizes[1] / 2;     // 800000
    const int Etot = E + N;

    // workspace layout (floats)
    float* ws   = (float*)d_ws;
    float* h1   = ws;                    // N*128
    float* out1 = h1   + (size_t)N * 128; // N*128  (becomes ELU features)
    float* as1  = out1 + (size_t)N * 128; // N*4
    float* ad1  = as1  + (size_t)N * 4;   // N*4
    float* m1   = ad1  + (size_t)N * 4;   // N*4
    float* den1 = m1   + (size_t)N * 4;   // N*4
    float* h2   = den1 + (size_t)N * 4;   // N*2
    float* as2  = h2   + (size_t)N * 2;   // N
    float* ad2  = as2  + (size_t)N;       // N
    float* m2   = ad2  + (size_t)N;       // N
    float* den2 = m2   + (size_t)N;       // N
    float* out2 = den2 + (size_t)N;       // N*2

    const int TB = 256;

    // init
    init_ws<<<(N * 128 + TB - 1) / TB, TB, 0, stream>>>(m1, den1, out1, m2, den2, out2, N);

    // layer 1 projection (WMMA) + attention scalars
    gemm_wmma_f32<<<N / 16, 256, 0, stream>>>(x, W1, h1, N);
    attn_l1<<<(N * 4 + TB - 1) / TB, TB, 0, stream>>>(h1, att_src1, att_dst1, as1, ad1, N);

    // layer 1 segment softmax + aggregation
    edge_max_l1<<<(Etot * 4 + TB - 1) / TB, TB, 0, stream>>>(ei, E, N, as1, ad1, m1);
    edge_sum_l1<<<(Etot * 4 + TB - 1) / TB, TB, 0, stream>>>(ei, E, N, as1, ad1, m1, den1);
    edge_agg_l1<<<(Etot + 7) / 8, 256, 0, stream>>>(ei, E, N, as1, ad1, m1, den1, h1, out1);
    bias_elu<<<(N * 128 + TB - 1) / TB, TB, 0, stream>>>(out1, b1, N);

    // layer 2 projection + attention scalars
    l2_matvec<<<(N + 7) / 8, 256, 0, stream>>>(out1, W2, att_src2, att_dst2, h2, as2, ad2, N);

    // layer 2 segment softmax + aggregation
    edge_max_l2<<<(Etot + TB - 1) / TB, TB, 0, stream>>>(ei, E, N, as2, ad2, m2);
    edge_sum_l2<<<(Etot + TB - 1) / TB, TB, 0, stream>>>(ei, E, N, as2, ad2, m2, den2);
    edge_agg_l2<<<(Etot + TB - 1) / TB, TB, 0, stream>>>(ei, E, N, as2, ad2, m2, den2, h2, out2);

    // bias + log_softmax
    final_logsoftmax<<<(N + TB - 1) / TB, TB, 0, stream>>>(out2, b2, y, N);
}